// GNN_5128190951839
// MI455X (gfx1250) — compile-verified
//
#include <hip/hip_runtime.h>

typedef __attribute__((ext_vector_type(2))) float v2f;
typedef __attribute__((ext_vector_type(8))) float v8f;

// ---------------------------------------------------------------------------
// Utility kernels
// ---------------------------------------------------------------------------
__global__ void zero_kernel(float4* __restrict__ p, int n4) {
  int i = blockIdx.x * blockDim.x + threadIdx.x;
  if (i < n4) {
    float4 z; z.x = 0.f; z.y = 0.f; z.z = 0.f; z.w = 0.f;
    p[i] = z;
  }
}

__global__ void deg_kernel(const int* __restrict__ src, const int* __restrict__ dst,
                           float* __restrict__ outdeg, float* __restrict__ indeg, int nedges) {
  int e = blockIdx.x * blockDim.x + threadIdx.x;
  if (e < nedges) {
    atomicAdd(&outdeg[src[e]], 1.0f);   // non-returning global_atomic_add_f32
    atomicAdd(&indeg[dst[e]], 1.0f);
  }
}

__global__ void norm_kernel(float* __restrict__ degs, int n) {
  int i = blockIdx.x * blockDim.x + threadIdx.x;
  if (i < n) degs[i] = rsqrtf(fmaxf(degs[i], 1.0f));
}

// Wt[c*K + k] = (c < NOUT) ? W[k*NOUT + c] : 0   (column-major, zero-padded)
template <int K, int NOUT, int NOUTP>
__global__ void transpose_w_kernel(const float* __restrict__ W, float* __restrict__ Wt) {
  int idx = blockIdx.x * blockDim.x + threadIdx.x;
  if (idx < NOUTP * K) {
    int c = idx / K;
    int k = idx % K;
    Wt[idx] = (c < NOUT) ? W[(size_t)k * NOUT + c] : 0.0f;
  }
}

// ---------------------------------------------------------------------------
// GEMM: Y[r, c] = norm[r] * sum_k X[r,k] * Wt[c,k]
// fp32 WMMA 16x16x4. One wave -> one 16(M) x 16(N) tile.
// blockDim.x = 32 * NTILES; blockIdx.x indexes 16-row slabs.
//
// VGPR layouts (CDNA5 ISA 7.12.2):
//   A 16x4 f32 : lanes 0-15 = rows, v0/v1 = K {k0,k0+1}; lanes 16-31: {k0+2,k0+3}
//   B 4x16 f32 : lanes 0-15 = cols, v0/v1 = K {k0,k0+1}; lanes 16-31: {k0+2,k0+3}
//   C/D 16x16  : VGPR j -> row j (lanes 0-15) / row j+8 (lanes 16-31), N = lane&15
//
// With column-major Wt the B fragment is a contiguous aligned 8-byte pair,
// symmetric with A: both operands are unconditional global_load_b64.
// ---------------------------------------------------------------------------
template <int K, int NTILES, int NSTORE>
__global__ void __launch_bounds__(128)
gemm_norm_kernel(const float* __restrict__ X, const float* __restrict__ Wt,
                 const float* __restrict__ norm, float* __restrict__ Y) {
  const int lane   = threadIdx.x & 31;
  const int wave   = threadIdx.x >> 5;
  const int laneLo = lane & 15;
  const int laneHi = lane >> 4;           // 0 or 1
  const int row0   = blockIdx.x * 16;     // 100000 % 16 == 0 -> always full
  const int n0     = wave * 16;

  const float* __restrict__ xrow = X + (size_t)(row0 + laneLo) * K + laneHi * 2;
  const float* __restrict__ wcol = Wt + (size_t)(n0 + laneLo) * K + laneHi * 2;

  v8f acc = {};
#pragma unroll
  for (int k0 = 0; k0 < K; k0 += 4) {
    v2f a = *(const v2f*)(xrow + k0);
    v2f b = *(const v2f*)(wcol + k0);
    acc = __builtin_amdgcn_wmma_f32_16x16x4_f32(false, a, false, b,
                                                (short)0, acc, false, false);
  }

  const int bcol = n0 + laneLo;
#pragma unroll
  for (int j = 0; j < 8; ++j) {
    const int r = row0 + j + laneHi * 8;
    if (NTILES * 16 <= NSTORE || bcol < NSTORE) {   // folds away for 64-wide layers
      Y[(size_t)r * NSTORE + bcol] = acc[j] * norm[r];
    }
  }
}

// ---------------------------------------------------------------------------
// Edge scatter: agg[dst[e], :] += xw[src[e], :]
// CHUNKS float4-chunks per edge (F = CHUNKS*4). Atomics resolve in L2
// (whole working set < 192 MB L2).
// ---------------------------------------------------------------------------
template <int CHUNKS>
__global__ void scatter_kernel(const int* __restrict__ src, const int* __restrict__ dst,
                               const float* __restrict__ xw, float* __restrict__ agg,
                               int nedges) {
  const int tid = blockIdx.x * blockDim.x + threadIdx.x;
  if (tid >= nedges * CHUNKS) return;
  const int e = tid / CHUNKS;
  const int c = tid % CHUNKS;
  const int s = src[e];
  const int d = dst[e];
  const float4 v = *(const float4*)(xw + (size_t)s * (CHUNKS * 4) + c * 4);
  float* p = agg + (size_t)d * (CHUNKS * 4) + c * 4;
  atomicAdd(p + 0, v.x);
  atomicAdd(p + 1, v.y);
  atomicAdd(p + 2, v.z);
  atomicAdd(p + 3, v.w);
}

// ---------------------------------------------------------------------------
// Finalize: out = (agg * norm_dst[row]) + bias, optional ReLU
// ---------------------------------------------------------------------------
template <int F, bool RELU>
__global__ void finalize_kernel(const float* __restrict__ agg, const float* __restrict__ norm,
                                const float* __restrict__ bias, float* __restrict__ out,
                                int nnodes) {
  const int tid = blockIdx.x * blockDim.x + threadIdx.x;
  if (tid >= nnodes * F) return;
  const int i = tid / F;
  const int f = tid % F;
  float v = agg[tid] * norm[i] + bias[f];
  out[tid] = RELU ? fmaxf(v, 0.0f) : v;
}

// ---------------------------------------------------------------------------
// Host orchestration
// ---------------------------------------------------------------------------
extern "C" void kernel_launch(void* const* d_in, const int* in_sizes, int n_in,
                              void* d_out, int out_size, void* d_ws, size_t ws_size,
                              hipStream_t stream) {
  const float* h  = (const float*)d_in[0];   // [N,128]
  const float* W0 = (const float*)d_in[1];   // [128,64]
  const float* b0 = (const float*)d_in[2];   // [64]
  const float* W1 = (const float*)d_in[3];   // [64,64]
  const float* b1 = (const float*)d_in[4];   // [64]
  const float* W2 = (const float*)d_in[5];   // [64,40]
  const float* b2 = (const float*)d_in[6];   // [40]
  const int* src  = (const int*)d_in[7];     // [E]
  const int* dst  = (const int*)d_in[8];     // [E]

  const int N = in_sizes[0] / 128;           // 100000
  const int E = in_sizes[7];                 // 1600000

  float* nsrc = (float*)d_ws;                // [N]
  float* ndst = nsrc + N;                    // [N]
  float* bufA = ndst + N;                    // [N*64]
  float* bufB = bufA + (size_t)N * 64;       // [N*64]
  float* Wt0  = bufB + (size_t)N * 64;       // [64*128]  col-major W0
  float* Wt1  = Wt0 + 64 * 128;              // [64*64]   col-major W1
  float* Wt2  = Wt1 + 64 * 64;               // [48*64]   col-major W2, padded 40->48
  float* outf = (float*)d_out;               // [N*40]

  const int BT = 256;
  auto blocks = [](int total, int bt) { return (total + bt - 1) / bt; };

  // --- weight transposition / padding (tiny, once per launch) ---
  transpose_w_kernel<128, 64, 64><<<blocks(64 * 128, BT), BT, 0, stream>>>(W0, Wt0);
  transpose_w_kernel<64, 64, 64><<<blocks(64 * 64, BT), BT, 0, stream>>>(W1, Wt1);
  transpose_w_kernel<64, 40, 48><<<blocks(48 * 64, BT), BT, 0, stream>>>(W2, Wt2);

  // --- degrees -> norms ---
  zero_kernel<<<blocks(2 * N / 4, BT), BT, 0, stream>>>((float4*)nsrc, 2 * N / 4);
  deg_kernel<<<blocks(E, BT), BT, 0, stream>>>(src, dst, nsrc, ndst, E);
  norm_kernel<<<blocks(2 * N, BT), BT, 0, stream>>>(nsrc, 2 * N);

  const int rowBlocks = N / 16;              // 6250, exact

  // --- layer 0: h[N,128] -> bufA(xw) -> bufB(agg) -> bufA(h1) ---
  gemm_norm_kernel<128, 4, 64><<<rowBlocks, 128, 0, stream>>>(h, Wt0, nsrc, bufA);
  zero_kernel<<<blocks(N * 64 / 4, BT), BT, 0, stream>>>((float4*)bufB, N * 64 / 4);
  scatter_kernel<16><<<blocks(E * 16, BT), BT, 0, stream>>>(src, dst, bufA, bufB, E);
  finalize_kernel<64, true><<<blocks(N * 64, BT), BT, 0, stream>>>(bufB, ndst, b0, bufA, N);

  // --- layer 1: bufA(h1) -> bufB(xw) -> bufA(agg) -> bufB(h2) ---
  gemm_norm_kernel<64, 4, 64><<<rowBlocks, 128, 0, stream>>>(bufA, Wt1, nsrc, bufB);
  zero_kernel<<<blocks(N * 64 / 4, BT), BT, 0, stream>>>((float4*)bufA, N * 64 / 4);
  scatter_kernel<16><<<blocks(E * 16, BT), BT, 0, stream>>>(src, dst, bufB, bufA, E);
  finalize_kernel<64, true><<<blocks(N * 64, BT), BT, 0, stream>>>(bufA, ndst, b1, bufB, N);

  // --- layer 2: bufB(h2) -> bufA(xw,40) -> bufB(agg,40) -> d_out ---
  gemm_norm_kernel<64, 3, 40><<<rowBlocks, 96, 0, stream>>>(bufB, Wt2, nsrc, bufA);
  zero_kernel<<<blocks(N * 40 / 4, BT), BT, 0, stream>>>((float4*)bufB, N * 40 / 4);
  scatter_kernel<10><<<blocks(E * 10, BT), BT, 0, stream>>>(src, dst, bufA, bufB, E);
  finalize_kernel<40, false><<<blocks(N * 40, BT), BT, 0, stream>>>(bufB, ndst, b2, outf, N);
}